// PlaneGraphEncoder_10737418240015
// MI455X (gfx1250) — compile-verified
//
#include <hip/hip_runtime.h>
#include <hip/hip_bf16.h>

#define NN 50000
#define EE 800000
#define GG 64
#define EPS 1e-5f

typedef __attribute__((ext_vector_type(2))) float v2f;
typedef __attribute__((ext_vector_type(8))) float v8f;

__device__ __forceinline__ float leaky(float x) { return x > 0.f ? x : 0.2f * x; }
__device__ __forceinline__ float bn1(float x, float g, float b, float m, float v) {
    return (x - m) * rsqrtf(v + EPS) * g + b;
}
__device__ __forceinline__ void atomicMaxF(float* addr, float v) {
    if (v >= 0.f) atomicMax((int*)addr, __float_as_int(v));
    else          atomicMin((unsigned int*)addr, __float_as_uint(v));
}
__device__ __forceinline__ float neg_inf() { return __int_as_float(0xFF800000); }

// ---- small precompute: kel[h]=dot(W1_h, al1_h), ker[h]=dot(W1_h, ar1_h) (IN==1) ----
__global__ void k_attproj1(const float* __restrict__ W1, const float* __restrict__ al1,
                           const float* __restrict__ ar1, float* __restrict__ kelker) {
    int h = threadIdx.x;
    if (h >= 2) return;
    float sl = 0.f, sr = 0.f;
    for (int d = 0; d < 64; ++d) {
        float w = W1[h * 64 + d];
        sl += w * al1[h * 64 + d];
        sr += w * ar1[h * 64 + d];
    }
    kelker[h] = sl;
    kelker[2 + h] = sr;
}

// ---- feat1[n,c] = feat[n]*W1[c]; zero msg accumulator ----
__global__ void k_featproj1(const float* __restrict__ feat, const float* __restrict__ W1,
                            float* __restrict__ feat1, float* __restrict__ acc1) {
    unsigned t = blockIdx.x * blockDim.x + threadIdx.x;
    if (t >= (unsigned)NN * 128u) return;
    unsigned n = t >> 7, c = t & 127u;
    feat1[t] = feat[n] * W1[c];
    acc1[t] = 0.f;
}

// ---- per-node el/er + softmax state init (both layers) ----
__global__ void k_node_prep(const float* __restrict__ feat, const float* __restrict__ kelker,
                            float* __restrict__ el1, float* __restrict__ er1,
                            float* __restrict__ m1, float* __restrict__ s1,
                            float* __restrict__ m2, float* __restrict__ s2) {
    unsigned n = blockIdx.x * blockDim.x + threadIdx.x;
    if (n >= NN) return;
    float f = feat[n];
    for (int h = 0; h < 2; ++h) {
        el1[n * 2 + h] = f * kelker[h];
        er1[n * 2 + h] = f * kelker[2 + h];
        m1[n * 2 + h] = neg_inf();
        s1[n * 2 + h] = 0.f;
    }
    m2[n] = neg_inf();
    s2[n] = 0.f;
}

__global__ void k_zero(float* __restrict__ p, unsigned n) {
    unsigned t = blockIdx.x * blockDim.x + threadIdx.x;
    if (t < n) p[t] = 0.f;
}

// ---- GAT1 edge pass 1: segment max ----
__global__ void k_edge_max1(const float* __restrict__ el1, const float* __restrict__ er1,
                            const int* __restrict__ src, const int* __restrict__ dst,
                            float* __restrict__ m1) {
    unsigned ei = blockIdx.x * blockDim.x + threadIdx.x;
    if (ei >= EE) return;
    int s = src[ei], d = dst[ei];
    for (int h = 0; h < 2; ++h) {
        float e = leaky(el1[s * 2 + h] + er1[d * 2 + h]);
        atomicMaxF(&m1[d * 2 + h], e);
    }
}

// ---- GAT1 edge pass 2: a = exp(e - m[dst]); s[dst]+=a; stash a ----
__global__ void k_edge_sum1(const float* __restrict__ el1, const float* __restrict__ er1,
                            const int* __restrict__ src, const int* __restrict__ dst,
                            const float* __restrict__ m1, float* __restrict__ s1,
                            float* __restrict__ a1) {
    unsigned ei = blockIdx.x * blockDim.x + threadIdx.x;
    if (ei >= EE) return;
    int s = src[ei], d = dst[ei];
    for (int h = 0; h < 2; ++h) {
        float e = leaky(el1[s * 2 + h] + er1[d * 2 + h]);
        float a = expf(e - m1[d * 2 + h]);
        a1[ei * 2 + h] = a;
        atomicAdd(&s1[d * 2 + h], a);
    }
}

// ---- GAT1 edge pass 3: message scatter, thread per (edge, channel) ----
__global__ void k_edge_msg1(const float* __restrict__ feat1, const float* __restrict__ a1,
                            const int* __restrict__ src, const int* __restrict__ dst,
                            float* __restrict__ acc1) {
    unsigned t = blockIdx.x * blockDim.x + threadIdx.x;
    if (t >= (unsigned)EE * 128u) return;
    unsigned ei = t >> 7, c = t & 127u;
    int s = src[ei], d = dst[ei];
    float a = a1[ei * 2 + (c >> 6)];
    atomicAdd(&acc1[(unsigned)d * 128u + c], feat1[(unsigned)s * 128u + c] * a);
}

// ---- GAT1 finalize: rst/s + residual + bias -> BN -> ReLU (in place acc1 -> h1) ----
__global__ void k_node_fin1(const float* __restrict__ feat, const float* __restrict__ res1,
                            const float* __restrict__ b1, const float* __restrict__ s1,
                            const float* __restrict__ gg, const float* __restrict__ gb,
                            const float* __restrict__ gm, const float* __restrict__ gv,
                            float* __restrict__ acc1) {
    unsigned t = blockIdx.x * blockDim.x + threadIdx.x;
    if (t >= (unsigned)NN * 128u) return;
    unsigned n = t >> 7, c = t & 127u;
    float sv = s1[n * 2 + (c >> 6)];
    float rst = sv > 0.f ? acc1[t] / sv : 0.f;
    float x = rst + feat[n] * res1[c] + b1[c];
    acc1[t] = fmaxf(bn1(x, gg[c], gb[c], gm[c], gv[c]), 0.f);
}

// ============================================================================
// GEMM2 via V_WMMA_F32_16X16X4_F32 with LDS-staged, WMMA-swizzled B operands.
//   feat2 = h1 @ W2, resid2 = h1 @ res2   (A = h1 tile [16x128], K in steps of 4)
// fp32 A(16x4) layout: lane<16 -> M=lane, K={k0,k0+1}; lane>=16 -> M=lane-16, K={k0+2,k0+3}
// B(4x16) layout mirrors A; C/D(16x16): vgpr r, lane l -> M = r + 8*(l>=16), N = l&15
// B fragments pre-swizzled into LDS as v2f[lane] -> one conflict-free ds_load_b64
// (32 lanes x 8B = 256B = all 64 banks exactly once).
// ============================================================================
__global__ __launch_bounds__(128) void k_gemm2(const float* __restrict__ h1,
                                               const float* __restrict__ W2,
                                               const float* __restrict__ Rw,
                                               float* __restrict__ feat2,
                                               float* __restrict__ resid2) {
    // Bsh index: ((mat*32 + kk)*2 + ntile)*32 + lane   (mat: 0=W2, 1=res2)
    __shared__ v2f Bsh[2 * 32 * 2 * 32];   // 32 KB
    unsigned lane = threadIdx.x & 31u;
    unsigned wave = threadIdx.x >> 5;

    // cooperative stage + swizzle: 4096 entries / 128 threads = 32 each
    for (unsigned e = threadIdx.x; e < 2u * 32u * 2u * 32u; e += 128u) {
        unsigned mat = e >> 11;
        unsigned r = e & 2047u;
        unsigned kk = r >> 6;
        unsigned nt = (r >> 5) & 1u;
        unsigned ln = r & 31u;
        unsigned kb = kk * 4u + 2u * (ln >> 4);
        unsigned n = nt * 16u + (ln & 15u);
        const float* srcm = mat ? Rw : W2;
        v2f val;
        val.x = srcm[kb * 32u + n];
        val.y = srcm[(kb + 1u) * 32u + n];
        Bsh[e] = val;
    }
    __syncthreads();

    unsigned tile = blockIdx.x * 4u + wave;
    if (tile >= NN / 16u) return;
    unsigned m0 = tile * 16u;
    unsigned half = lane >> 4, mr = lane & 15u;
    const float* arow = h1 + (size_t)(m0 + mr) * 128u;

    v8f aW0 = {}, aW1 = {}, aR0 = {}, aR1 = {};
    for (int kk = 0; kk < 32; ++kk) {
        int kb = kk * 4 + 2 * (int)half;
        v2f A;
        A.x = arow[kb];
        A.y = arow[kb + 1];
        v2f B0 = Bsh[((0 * 32 + kk) * 2 + 0) * 32 + lane];
        v2f B1 = Bsh[((0 * 32 + kk) * 2 + 1) * 32 + lane];
        v2f C0 = Bsh[((1 * 32 + kk) * 2 + 0) * 32 + lane];
        v2f C1 = Bsh[((1 * 32 + kk) * 2 + 1) * 32 + lane];
        aW0 = __builtin_amdgcn_wmma_f32_16x16x4_f32(false, A, false, B0, (short)0, aW0, false, false);
        aW1 = __builtin_amdgcn_wmma_f32_16x16x4_f32(false, A, false, B1, (short)0, aW1, false, false);
        aR0 = __builtin_amdgcn_wmma_f32_16x16x4_f32(false, A, false, C0, (short)0, aR0, false, false);
        aR1 = __builtin_amdgcn_wmma_f32_16x16x4_f32(false, A, false, C1, (short)0, aR1, false, false);
    }
    for (int r = 0; r < 8; ++r) {
        unsigned m = m0 + (unsigned)r + 8u * half;
        feat2[m * 32 + mr]       = aW0[r];
        feat2[m * 32 + 16 + mr]  = aW1[r];
        resid2[m * 32 + mr]      = aR0[r];
        resid2[m * 32 + 16 + mr] = aR1[r];
    }
}

// ---- el2/er2 per node ----
__global__ void k_elr2(const float* __restrict__ feat2, const float* __restrict__ al2,
                       const float* __restrict__ ar2, float* __restrict__ el2,
                       float* __restrict__ er2) {
    unsigned n = blockIdx.x * blockDim.x + threadIdx.x;
    if (n >= NN) return;
    float sl = 0.f, sr = 0.f;
    const float* f = feat2 + (size_t)n * 32u;
    for (int c = 0; c < 32; ++c) {
        sl += f[c] * al2[c];
        sr += f[c] * ar2[c];
    }
    el2[n] = sl;
    er2[n] = sr;
}

__global__ void k_edge_max2(const float* __restrict__ el2, const float* __restrict__ er2,
                            const int* __restrict__ src, const int* __restrict__ dst,
                            float* __restrict__ m2) {
    unsigned ei = blockIdx.x * blockDim.x + threadIdx.x;
    if (ei >= EE) return;
    float e = leaky(el2[src[ei]] + er2[dst[ei]]);
    atomicMaxF(&m2[dst[ei]], e);
}

__global__ void k_edge_sum2(const float* __restrict__ el2, const float* __restrict__ er2,
                            const int* __restrict__ src, const int* __restrict__ dst,
                            const float* __restrict__ m2, float* __restrict__ s2,
                            float* __restrict__ a2) {
    unsigned ei = blockIdx.x * blockDim.x + threadIdx.x;
    if (ei >= EE) return;
    int d = dst[ei];
    float e = leaky(el2[src[ei]] + er2[d]);
    float a = expf(e - m2[d]);
    a2[ei] = a;
    atomicAdd(&s2[d], a);
}

__global__ void k_edge_msg2(const float* __restrict__ feat2, const float* __restrict__ a2,
                            const int* __restrict__ src, const int* __restrict__ dst,
                            float* __restrict__ acc2) {
    unsigned t = blockIdx.x * blockDim.x + threadIdx.x;
    if (t >= (unsigned)EE * 32u) return;
    unsigned ei = t >> 5, c = t & 31u;
    atomicAdd(&acc2[(unsigned)dst[ei] * 32u + c], feat2[(unsigned)src[ei] * 32u + c] * a2[ei]);
}

__global__ void k_node_fin2(const float* __restrict__ resid2, const float* __restrict__ b2,
                            const float* __restrict__ s2, const float* __restrict__ gg,
                            const float* __restrict__ gb, const float* __restrict__ gm,
                            const float* __restrict__ gv, float* __restrict__ acc2) {
    unsigned t = blockIdx.x * blockDim.x + threadIdx.x;
    if (t >= (unsigned)NN * 32u) return;
    unsigned n = t >> 5, c = t & 31u;
    float sv = s2[n];
    float rst = sv > 0.f ? acc2[t] / sv : 0.f;
    float x = rst + resid2[t] + b2[c];
    acc2[t] = fmaxf(bn1(x, gg[c], gb[c], gm[c], gv[c]), 0.f);
}

// ---- graph mean pooling ----
__global__ void k_graph_acc(const float* __restrict__ h2, const int* __restrict__ gid,
                            float* __restrict__ gsum, float* __restrict__ gcnt) {
    unsigned t = blockIdx.x * blockDim.x + threadIdx.x;
    if (t >= (unsigned)NN * 32u) return;
    unsigned n = t >> 5, c = t & 31u;
    int g = gid[n];
    atomicAdd(&gsum[g * 32 + c], h2[t]);
    if (c == 0) atomicAdd(&gcnt[g], 1.f);
}

__global__ void k_graph_out(const float* __restrict__ gsum, const float* __restrict__ gcnt,
                            float* __restrict__ out) {
    unsigned t = blockIdx.x * blockDim.x + threadIdx.x;
    if (t >= (unsigned)GG * 32u) return;
    out[t] = gsum[t] / fmaxf(gcnt[t >> 5], 1.f);
}

// ============================================================================
// Decoder layer 1 via WMMA: d = relu(bn(h2 @ Wd1 + bd1)), K=32, Nout=128.
// Wd1 LDS-staged & swizzled; A fragments (kk-only dependent) preloaded once
// into 8 v2f registers and reused across all 8 N-tiles.
// ============================================================================
__global__ __launch_bounds__(128) void k_dec1(const float* __restrict__ h2,
                                              const float* __restrict__ Wd1,
                                              const float* __restrict__ bd1,
                                              const float* __restrict__ gg,
                                              const float* __restrict__ gb,
                                              const float* __restrict__ gm,
                                              const float* __restrict__ gv,
                                              float* __restrict__ dbuf) {
    // Wsh index: (kk*8 + nt)*32 + lane  -> 2048 v2f = 16 KB
    __shared__ v2f Wsh[8 * 8 * 32];
    unsigned lane = threadIdx.x & 31u;
    unsigned wave = threadIdx.x >> 5;

    for (unsigned e = threadIdx.x; e < 8u * 8u * 32u; e += 128u) {
        unsigned kk = e >> 8;
        unsigned nt = (e >> 5) & 7u;
        unsigned ln = e & 31u;
        unsigned kb = kk * 4u + 2u * (ln >> 4);
        unsigned n = nt * 16u + (ln & 15u);
        v2f val;
        val.x = Wd1[kb * 128u + n];
        val.y = Wd1[(kb + 1u) * 128u + n];
        Wsh[e] = val;
    }
    __syncthreads();

    unsigned tile = blockIdx.x * 4u + wave;
    if (tile >= NN / 16u) return;
    unsigned m0 = tile * 16u;
    unsigned half = lane >> 4, mr = lane & 15u;
    const float* arow = h2 + (size_t)(m0 + mr) * 32u;

    // preload all A fragments (K=32 -> 8 k-steps)
    v2f Areg[8];
#pragma unroll
    for (int kk = 0; kk < 8; ++kk) {
        int kb = kk * 4 + 2 * (int)half;
        Areg[kk].x = arow[kb];
        Areg[kk].y = arow[kb + 1];
    }

    for (int nt = 0; nt < 8; ++nt) {
        unsigned n = nt * 16 + mr;
        v8f acc = {};
#pragma unroll
        for (int kk = 0; kk < 8; ++kk) {
            v2f B = Wsh[(kk * 8 + nt) * 32 + lane];
            acc = __builtin_amdgcn_wmma_f32_16x16x4_f32(false, Areg[kk], false, B, (short)0, acc, false, false);
        }
        float sc = rsqrtf(gv[n] + EPS) * gg[n];
        float mu = gm[n], be = gb[n], bi = bd1[n];
        for (int r = 0; r < 8; ++r) {
            unsigned m = m0 + (unsigned)r + 8u * half;
            float x = (acc[r] + bi - mu) * sc + be;
            dbuf[(size_t)m * 128u + n] = fmaxf(x, 0.f);
        }
    }
}

// ---- decoder layer 2: recon[n] = d[n,:] . Wd2 + bd2 ----
__global__ void k_dec2(const float* __restrict__ dbuf, const float* __restrict__ Wd2,
                       const float* __restrict__ bd2, float* __restrict__ out) {
    unsigned n = blockIdx.x * blockDim.x + threadIdx.x;
    if (n >= NN) return;
    const float* r = dbuf + (size_t)n * 128u;
    float acc = 0.f;
#pragma unroll 8
    for (int c = 0; c < 128; ++c) acc += r[c] * Wd2[c];
    out[n] = acc + bd2[0];
}

static inline unsigned cdiv(unsigned a, unsigned b) { return (a + b - 1) / b; }

extern "C" void kernel_launch(void* const* d_in, const int* in_sizes, int n_in,
                              void* d_out, int out_size, void* d_ws, size_t ws_size,
                              hipStream_t stream) {
    const float* feat = (const float*)d_in[0];
    const float* W1   = (const float*)d_in[1];
    const float* al1  = (const float*)d_in[2];
    const float* ar1  = (const float*)d_in[3];
    const float* res1 = (const float*)d_in[4];
    const float* b1   = (const float*)d_in[5];
    const float* g1g  = (const float*)d_in[6];
    const float* g1b  = (const float*)d_in[7];
    const float* g1m  = (const float*)d_in[8];
    const float* g1v  = (const float*)d_in[9];
    const float* W2   = (const float*)d_in[10];
    const float* al2  = (const float*)d_in[11];
    const float* ar2  = (const float*)d_in[12];
    const float* res2 = (const float*)d_in[13];
    const float* b2   = (const float*)d_in[14];
    const float* g2g  = (const float*)d_in[15];
    const float* g2b  = (const float*)d_in[16];
    const float* g2m  = (const float*)d_in[17];
    const float* g2v  = (const float*)d_in[18];
    const float* Wd1  = (const float*)d_in[19];
    const float* bd1  = (const float*)d_in[20];
    const float* gdg  = (const float*)d_in[21];
    const float* gdb  = (const float*)d_in[22];
    const float* gdm  = (const float*)d_in[23];
    const float* gdv  = (const float*)d_in[24];
    const float* Wd2  = (const float*)d_in[25];
    const float* bd2  = (const float*)d_in[26];
    const int*   src  = (const int*)d_in[27];
    const int*   dst  = (const int*)d_in[28];
    const int*   gid  = (const int*)d_in[29];
    float* out = (float*)d_out;

    // workspace carve-out (floats)
    float* w = (float*)d_ws;
    size_t o = 0;
    float* feat1  = w + o; o += (size_t)NN * 128;
    float* acc1   = w + o; o += (size_t)NN * 128;   // becomes h1
    float* el1    = w + o; o += (size_t)NN * 2;
    float* er1    = w + o; o += (size_t)NN * 2;
    float* m1     = w + o; o += (size_t)NN * 2;
    float* s1     = w + o; o += (size_t)NN * 2;
    float* a1     = w + o; o += (size_t)EE * 2;
    float* feat2  = w + o; o += (size_t)NN * 32;
    float* resid2 = w + o; o += (size_t)NN * 32;
    float* el2    = w + o; o += (size_t)NN;
    float* er2    = w + o; o += (size_t)NN;
    float* m2     = w + o; o += (size_t)NN;
    float* s2     = w + o; o += (size_t)NN;
    float* a2     = w + o; o += (size_t)EE;
    float* acc2   = w + o; o += (size_t)NN * 32;    // becomes h2
    float* dbuf   = w + o; o += (size_t)NN * 128;
    float* gsum   = w + o; o += (size_t)GG * 32;
    float* gcnt   = w + o; o += (size_t)GG;
    float* kelker = w + o; o += 4;

    const unsigned B = 256;

    // layer 1 prep
    k_attproj1<<<1, 32, 0, stream>>>(W1, al1, ar1, kelker);
    k_featproj1<<<cdiv(NN * 128u, B), B, 0, stream>>>(feat, W1, feat1, acc1);
    k_node_prep<<<cdiv(NN, B), B, 0, stream>>>(feat, kelker, el1, er1, m1, s1, m2, s2);
    k_zero<<<cdiv(NN * 32u, B), B, 0, stream>>>(acc2, NN * 32u);
    k_zero<<<cdiv(GG * 33u, B), B, 0, stream>>>(gsum, GG * 33u);  // gsum + gcnt contiguous

    // GAT layer 1
    k_edge_max1<<<cdiv(EE, B), B, 0, stream>>>(el1, er1, src, dst, m1);
    k_edge_sum1<<<cdiv(EE, B), B, 0, stream>>>(el1, er1, src, dst, m1, s1, a1);
    k_edge_msg1<<<cdiv(EE * 128u, B), B, 0, stream>>>(feat1, a1, src, dst, acc1);
    k_node_fin1<<<cdiv(NN * 128u, B), B, 0, stream>>>(feat, res1, b1, s1, g1g, g1b, g1m, g1v, acc1);

    // GAT layer 2 projection via WMMA (feat2 = h1@W2, resid2 = h1@res2)
    k_gemm2<<<cdiv(NN / 16u, 4u), 128, 0, stream>>>(acc1, W2, res2, feat2, resid2);
    k_elr2<<<cdiv(NN, B), B, 0, stream>>>(feat2, al2, ar2, el2, er2);
    k_edge_max2<<<cdiv(EE, B), B, 0, stream>>>(el2, er2, src, dst, m2);
    k_edge_sum2<<<cdiv(EE, B), B, 0, stream>>>(el2, er2, src, dst, m2, s2, a2);
    k_edge_msg2<<<cdiv(EE * 32u, B), B, 0, stream>>>(feat2, a2, src, dst, acc2);
    k_node_fin2<<<cdiv(NN * 32u, B), B, 0, stream>>>(resid2, b2, s2, g2g, g2b, g2m, g2v, acc2);

    // graph mean pooling -> out[0 .. G*32)
    k_graph_acc<<<cdiv(NN * 32u, B), B, 0, stream>>>(acc2, gid, gsum, gcnt);
    k_graph_out<<<cdiv(GG * 32u, B), B, 0, stream>>>(gsum, gcnt, out);

    // decoder -> out[G*32 .. G*32+N)
    k_dec1<<<cdiv(NN / 16u, 4u), 128, 0, stream>>>(acc2, Wd1, bd1, gdg, gdb, gdm, gdv, dbuf);
    k_dec2<<<cdiv(NN, B), B, 0, stream>>>(dbuf, Wd2, bd2, out + GG * 32);
}